// OutlookAttention_2302102471183
// MI455X (gfx1250) — compile-verified
//
#include <hip/hip_runtime.h>
#include <hip/hip_bf16.h>
#include <math.h>

typedef float v2f __attribute__((ext_vector_type(2)));
typedef float v8f __attribute__((ext_vector_type(8)));

#define HEADS 6
#define CDIM  192
#define HD    32
#define K2    9
#define HH    56
#define WW    56
#define HWP   (HH * WW)          // 3136
#define BATCH 8
#define ROWS  (BATCH * HWP)      // 25088  (divisible by 64)
#define K4H   (K2 * K2 * HEADS)  // 486

// ---------------------------------------------------------------------------
// fp32 WMMA GEMM: C[M,N] = A[M,KC] * B[KC,N] (+ bias[N] if non-null)
// Each wave32 computes a 64x16 C tile (4 M-subtiles) -> 4 V_WMMA_F32_16X16X4_F32
// per k-stage sharing one B fragment. KC is a template parameter so the
// k-loop fully unrolls; fragments live in a 3-slot rotating buffer with
// compile-time indices: stage s loads slot s%3 while stage s-2 (slot (s+1)%3)
// is consumed, so every WMMA's operands were loaded 2 stages (12 loads)
// earlier -> waits retire with outstanding loads still in flight.
// Fragment layouts per CDNA5 ISA 7.12.2:
//   A 16x4 : lane holds A[lane&15][2*(lane>>4) + {0,1}] in {v0,v1}
//   B 4x16 : lane holds B[2*(lane>>4) + {0,1}][lane&15] in {v0,v1}
//   C/D    : VGPR r -> row (r + 8*(lane>>4)), col (lane&15)
// OOB N columns: B column index clamped (dup loads of col N-1); garbage only
// reaches accumulator lanes whose store is guarded by ocol < N.
// ---------------------------------------------------------------------------
template <int KC>
__global__ __launch_bounds__(256) void gemm_f32_wmma(
    const float* __restrict__ A, const float* __restrict__ Bm,
    const float* __restrict__ bias, float* __restrict__ C,
    int M, int N)
{
  constexpr int NS = KC / 4;                 // 48 k-stages
  const int lane = threadIdx.x & 31;
  const int wave = threadIdx.x >> 5;
  const int ntn  = (N + 15) >> 4;
  const int ntm  = M >> 6;                   // 64-row blocks (M % 64 == 0)
  const int ntiles = ntm * ntn;
  const int tile = blockIdx.x * 8 + wave;
  if (tile >= ntiles) return;                // whole wave exits together
  const int tm = tile / ntn;
  const int tn = tile % ntn;

  const int m16 = lane & 15;
  const int kh  = (lane >> 4) << 1;          // 0 or 2

  // B fragment pointer (clamped column), advanced by 4*N per k-stage
  const int bcol  = tn * 16 + m16;
  const int bcolc = bcol < N ? bcol : N - 1;
  const float* Bp = Bm + (size_t)kh * N + bcolc;
  const size_t bstep = 4 * (size_t)N;

  // A fragment pointers for the 4 M-subtiles (8B-aligned: kh even, KC even)
  const int row0 = tm * 64 + m16;
  const float* Ap0 = A + (size_t)(row0     ) * KC + kh;
  const float* Ap1 = A + (size_t)(row0 + 16) * KC + kh;
  const float* Ap2 = A + (size_t)(row0 + 32) * KC + kh;
  const float* Ap3 = A + (size_t)(row0 + 48) * KC + kh;

  v8f acc0 = {0.f,0.f,0.f,0.f,0.f,0.f,0.f,0.f};
  v8f acc1 = acc0, acc2 = acc0, acc3 = acc0;

  // 3-slot rotating fragment buffers (indices constant after full unroll)
  v2f aF0[3], aF1[3], aF2[3], aF3[3], bF[3];

  // prologue: stages 0 and 1 into slots 0 and 1
  bF[0].x = Bp[0]; bF[0].y = Bp[N];
  aF0[0] = *(const v2f*)(Ap0);
  aF1[0] = *(const v2f*)(Ap1);
  aF2[0] = *(const v2f*)(Ap2);
  aF3[0] = *(const v2f*)(Ap3);
  Bp += bstep;
  bF[1].x = Bp[0]; bF[1].y = Bp[N];
  aF0[1] = *(const v2f*)(Ap0 + 4);
  aF1[1] = *(const v2f*)(Ap1 + 4);
  aF2[1] = *(const v2f*)(Ap2 + 4);
  aF3[1] = *(const v2f*)(Ap3 + 4);

  #pragma unroll
  for (int s = 2; s < NS; ++s) {
    const int ld = s % 3;                    // slot receiving stage s
    const int cu = (s + 1) % 3;              // slot holding stage s-2

    // issue stage s loads (A offsets are compile-time immediates)
    Bp += bstep;
    bF[ld].x = Bp[0]; bF[ld].y = Bp[N];
    aF0[ld] = *(const v2f*)(Ap0 + 4 * s);
    aF1[ld] = *(const v2f*)(Ap1 + 4 * s);
    aF2[ld] = *(const v2f*)(Ap2 + 4 * s);
    aF3[ld] = *(const v2f*)(Ap3 + 4 * s);

    // consume stage s-2 (its loads were issued 2 stages ago)
    acc0 = __builtin_amdgcn_wmma_f32_16x16x4_f32(false, aF0[cu], false, bF[cu], (short)0, acc0, false, false);
    acc1 = __builtin_amdgcn_wmma_f32_16x16x4_f32(false, aF1[cu], false, bF[cu], (short)0, acc1, false, false);
    acc2 = __builtin_amdgcn_wmma_f32_16x16x4_f32(false, aF2[cu], false, bF[cu], (short)0, acc2, false, false);
    acc3 = __builtin_amdgcn_wmma_f32_16x16x4_f32(false, aF3[cu], false, bF[cu], (short)0, acc3, false, false);
  }

  // epilogue: consume stages NS-2 and NS-1
  {
    const int c0 = (NS - 2) % 3;
    const int c1 = (NS - 1) % 3;
    acc0 = __builtin_amdgcn_wmma_f32_16x16x4_f32(false, aF0[c0], false, bF[c0], (short)0, acc0, false, false);
    acc1 = __builtin_amdgcn_wmma_f32_16x16x4_f32(false, aF1[c0], false, bF[c0], (short)0, acc1, false, false);
    acc2 = __builtin_amdgcn_wmma_f32_16x16x4_f32(false, aF2[c0], false, bF[c0], (short)0, acc2, false, false);
    acc3 = __builtin_amdgcn_wmma_f32_16x16x4_f32(false, aF3[c0], false, bF[c0], (short)0, acc3, false, false);
    acc0 = __builtin_amdgcn_wmma_f32_16x16x4_f32(false, aF0[c1], false, bF[c1], (short)0, acc0, false, false);
    acc1 = __builtin_amdgcn_wmma_f32_16x16x4_f32(false, aF1[c1], false, bF[c1], (short)0, acc1, false, false);
    acc2 = __builtin_amdgcn_wmma_f32_16x16x4_f32(false, aF2[c1], false, bF[c1], (short)0, acc2, false, false);
    acc3 = __builtin_amdgcn_wmma_f32_16x16x4_f32(false, aF3[c1], false, bF[c1], (short)0, acc3, false, false);
  }

  // stores: per VGPR r, lanes 0-15 -> row r, lanes 16-31 -> row r+8
  const int ocol = tn * 16 + m16;
  if (ocol < N) {
    const float bv = bias ? bias[ocol] : 0.f;
    const int orow0 = tm * 64 + ((lane >> 4) << 3);
    float* Cp = C + (size_t)orow0 * N + ocol;
    #pragma unroll
    for (int r = 0; r < 8; ++r) {
      Cp[(size_t)(r     ) * N] = acc0[r] + bv;
      Cp[(size_t)(r + 16) * N] = acc1[r] + bv;
      Cp[(size_t)(r + 32) * N] = acc2[r] + bv;
      Cp[(size_t)(r + 48) * N] = acc3[r] + bv;
    }
  }
}

// ---------------------------------------------------------------------------
// Softmax over contiguous groups of 9 (per b,pixel,head,k-row), scale = 32^-0.5
// ---------------------------------------------------------------------------
__global__ __launch_bounds__(256) void softmax9_kernel(float* __restrict__ attn,
                                                       int ngroups)
{
  const int g = blockIdx.x * blockDim.x + threadIdx.x;
  if (g >= ngroups) return;
  const int row = g / (HEADS * K2);
  const int grp = g % (HEADS * K2);
  float* p = attn + (size_t)row * K4H + grp * K2;

  const float scale = 0.17677669529663687f;  // 1/sqrt(32)
  float v[K2];
  float m = -INFINITY;
  #pragma unroll
  for (int i = 0; i < K2; ++i) { v[i] = p[i] * scale; m = fmaxf(m, v[i]); }
  float s = 0.f;
  #pragma unroll
  for (int i = 0; i < K2; ++i) { v[i] = __expf(v[i] - m); s += v[i]; }
  const float inv = 1.f / s;
  #pragma unroll
  for (int i = 0; i < K2; ++i) p[i] = v[i] * inv;
}

// ---------------------------------------------------------------------------
// Fused (unfold -> attn @ v -> fold): one wave per (b, pixel, head),
// lane = channel d (hd == 32 == wave32). Each lane caches the 5x5 v
// neighborhood in registers; fold means output pixel p sums over the 9
// window-centers c = p + (1-i, 1-j) that contain it, each contributing
// sum_q a[c, k=(i,j), q] * v[c + (qi-1, qj-1)].
// ---------------------------------------------------------------------------
__global__ __launch_bounds__(256) void attn_apply_fold_kernel(
    const float* __restrict__ attn, const float* __restrict__ v,
    float* __restrict__ y)
{
  const int wid = blockIdx.x * 8 + (threadIdx.x >> 5);
  const int total = BATCH * HWP * HEADS;
  if (wid >= total) return;
  const int head = wid % HEADS;
  const int bp   = wid / HEADS;
  const int p    = bp % HWP;
  const int b    = bp / HWP;
  const int pr   = p / WW;
  const int pc   = p % WW;
  const int d    = threadIdx.x & 31;

  // preload v[p-2 .. p+2]^2 for this (b, head, d); zero outside image (pad)
  float vn[25];
  #pragma unroll
  for (int dy = -2; dy <= 2; ++dy) {
    #pragma unroll
    for (int dx = -2; dx <= 2; ++dx) {
      const int r = pr + dy, c = pc + dx;
      float val = 0.f;
      if (r >= 0 && r < HH && c >= 0 && c < WW)
        val = v[(size_t)(b * HWP + r * WW + c) * CDIM + head * HD + d];
      vn[(dy + 2) * 5 + (dx + 2)] = val;
    }
  }

  float acc = 0.f;
  #pragma unroll
  for (int i = 0; i < 3; ++i) {
    #pragma unroll
    for (int j = 0; j < 3; ++j) {
      const int cr = pr + 1 - i, cc = pc + 1 - j;
      if (cr < 0 || cr >= HH || cc < 0 || cc >= WW) continue;
      const float* arow = attn + (size_t)(b * HWP + cr * WW + cc) * K4H
                               + (head * K2 + (i * 3 + j)) * K2;
      #pragma unroll
      for (int qi = 0; qi < 3; ++qi) {
        #pragma unroll
        for (int qj = 0; qj < 3; ++qj) {
          const int ny = qi - i, nx = qj - j;      // in [-2, 2]
          acc = fmaf(arow[qi * 3 + qj], vn[(ny + 2) * 5 + (nx + 2)], acc);
        }
      }
    }
  }
  y[(size_t)(b * HWP + p) * CDIM + head * HD + d] = acc;
}

// ---------------------------------------------------------------------------
extern "C" void kernel_launch(void* const* d_in, const int* in_sizes, int n_in,
                              void* d_out, int out_size, void* d_ws, size_t ws_size,
                              hipStream_t stream)
{
  const float* x  = (const float*)d_in[0];
  const float* Wv = (const float*)d_in[1];
  const float* Wa = (const float*)d_in[2];
  const float* ba = (const float*)d_in[3];
  const float* Wp = (const float*)d_in[4];
  const float* bp = (const float*)d_in[5];
  float* out = (float*)d_out;

  float* v    = (float*)d_ws;                       // ROWS*CDIM  (19.3 MB)
  float* attn = v    + (size_t)ROWS * CDIM;         // ROWS*K4H   (48.8 MB)
  float* y    = attn + (size_t)ROWS * K4H;          // ROWS*CDIM  (19.3 MB)

  // 1) v = x @ Wv   (tiles: (25088/64) * 12 = 4704 waves)
  {
    const int ntiles = (ROWS / 64) * ((CDIM + 15) / 16);
    gemm_f32_wmma<CDIM><<<(ntiles + 7) / 8, 256, 0, stream>>>(
        x, Wv, nullptr, v, ROWS, CDIM);
  }
  // 2) attn = x @ Wa + ba   (tiles: 392 * 31 = 12152 waves)
  {
    const int ntiles = (ROWS / 64) * ((K4H + 15) / 16);
    gemm_f32_wmma<CDIM><<<(ntiles + 7) / 8, 256, 0, stream>>>(
        x, Wa, ba, attn, ROWS, K4H);
  }
  // 3) softmax over groups of 9
  {
    const int ngroups = ROWS * HEADS * K2;                       // 1,354,752
    softmax9_kernel<<<(ngroups + 255) / 256, 256, 0, stream>>>(attn, ngroups);
  }
  // 4) fused unfold + attn@v + fold
  {
    const int nwaves = BATCH * HWP * HEADS;                      // 150,528
    attn_apply_fold_kernel<<<(nwaves + 7) / 8, 256, 0, stream>>>(attn, v, y);
  }
  // 5) out = y @ Wp + bp
  {
    const int ntiles = (ROWS / 64) * ((CDIM + 15) / 16);
    gemm_f32_wmma<CDIM><<<(ntiles + 7) / 8, 256, 0, stream>>>(
        y, Wp, bp, out, ROWS, CDIM);
  }
}